// StateClass_39805756900094
// MI455X (gfx1250) — compile-verified
//
#include <hip/hip_runtime.h>
#include <math.h>

#define BLK 256
#define HALO_PTS (BLK + 4)   // 260 points per tile: [base-2, base+257]
#define HINGES_T (BLK + 2)   // 258 hinge slots per tile: [base-2, base+255]
#define NWAVES (BLK / 32)

// ---------------------------------------------------------------------------
// One thread per point i. Stage the position tile into LDS with the CDNA5
// async global->LDS DMA (clamped addresses => no divergence), compute each
// hinge's angle gradient exactly once into LDS, then assemble per-point
// forces. Energies reduced wave-locally (shfl) then per-block -> d_ws.
// ---------------------------------------------------------------------------
__global__ __launch_bounds__(BLK) void chain_force_kernel(
    const float* __restrict__ pos,       // (N+2)*2
    const float* __restrict__ th_ss,     // N
    const float* __restrict__ rest,      // N+1
    const float* __restrict__ kstiff_p,
    const float* __restrict__ ksoft_p,
    const float* __restrict__ kstretch_p,
    const int2*  __restrict__ buckle,    // N pairs
    float* __restrict__ out,             // 3 + 2*(N+2)
    float* __restrict__ blockpart,       // 2*gridDim or nullptr
    int N)                               // hinge count
{
    __shared__ __align__(16) float sp[HALO_PTS * 2];  // position tile
    __shared__ float sg[HINGES_T * 4];                // G1x,G1y,G2x,G2y per hinge
    __shared__ float swr[NWAVES];
    __shared__ float sws[NWAVES];

    const int t    = threadIdx.x;
    const int base = blockIdx.x * BLK;   // point index of thread 0
    const int NP   = N + 2;

    // ---- Phase 0: async DMA of pos tile into LDS (8 B = one point per lane).
    // Clamped indices: OOB slots hold harmless duplicates and are never read.
    {
        int gp = min(max(base - 2 + t, 0), NP - 1);
        unsigned lds_off = (unsigned)(size_t)(&sp[2 * t]);
        unsigned long long ga = (unsigned long long)(size_t)(pos + 2 * (size_t)gp);
        asm volatile("global_load_async_to_lds_b64 %0, %1, off"
                     :: "v"(lds_off), "v"(ga) : "memory");
        if (t < HALO_PTS - BLK) {                    // 4 tail slots
            int s2 = t + BLK;
            gp = min(max(base - 2 + s2, 0), NP - 1);
            lds_off = (unsigned)(size_t)(&sp[2 * s2]);
            ga = (unsigned long long)(size_t)(pos + 2 * (size_t)gp);
            asm volatile("global_load_async_to_lds_b64 %0, %1, off"
                         :: "v"(lds_off), "v"(ga) : "memory");
        }
    }
    asm volatile("s_wait_asynccnt 0x0" ::: "memory");
    __syncthreads();

    const float kstiff   = *kstiff_p;
    const float ksoft    = *ksoft_p;
    const float kstretch = *kstretch_p;

    // ---- Phase 1: per-hinge theta + gradient terms (each hinge once).
    float rot_p = 0.0f;
    #pragma unroll
    for (int step = 0; step < 2; ++step) {
        int h = t + step * BLK;          // slots 0..255, then 256..257 (t<2)
        if (step == 1 && t >= HINGES_T - BLK) break;
        int j = base - 2 + h;            // hinge index
        if (j >= 0 && j < N) {
            float ax = sp[2*h+0], ay = sp[2*h+1];
            float bx = sp[2*h+2], by = sp[2*h+3];
            float cx = sp[2*h+4], cy = sp[2*h+5];
            float v1x = bx - ax, v1y = by - ay;
            float v2x = cx - bx, v2y = cy - by;
            float cr = v1x * v2y - v1y * v2x;
            float dt = v1x * v2x + v1y * v2y;
            float T  = atan2f(cr, dt);
            float TH = th_ss[j];
            int2  b  = buckle[j];
            float k0 = ((b.x == 1) ? (T > -TH) : (T < TH)) ? kstiff : ksoft;
            float k1 = ((b.y == 1) ? (T > -TH) : (T < TH)) ? kstiff : ksoft;
            float K  = k0 + k1;
            float d  = T - TH;
            float inv_r2 = 1.0f / (cr * cr + dt * dt);
            float gi = (K * d) * inv_r2;           // dE/dtheta / r^2
            sg[4*h+0] = gi * ( dt * v2y - cr * v2x);   // g * dtheta/dv1
            sg[4*h+1] = gi * (-dt * v2x - cr * v2y);
            sg[4*h+2] = gi * (-dt * v1y - cr * v1x);   // g * dtheta/dv2
            sg[4*h+3] = gi * ( dt * v1x - cr * v1y);
            if (h >= 2) rot_p += 0.5f * K * d * d;     // own hinge (j >= base)
        }
    }
    __syncthreads();

    // ---- Phase 2: per-point force assembly + stretch energy.
    const int i = base + t;              // point index (tile slot t+2)
    float str_p = 0.0f;
    if (i < NP) {
        float gx = 0.0f, gy = 0.0f;      // gradient accumulator
        // hinge j=i, role a: grad += -G1
        if (i < N)               { gx -= sg[4*(t+2)+0]; gy -= sg[4*(t+2)+1]; }
        // hinge j=i-1, role b: grad += G1 - G2
        if (i >= 1 && i - 1 < N) { gx += sg[4*(t+1)+0] - sg[4*(t+1)+2];
                                   gy += sg[4*(t+1)+1] - sg[4*(t+1)+3]; }
        // hinge j=i-2, role c: grad += +G2
        if (i >= 2 && i - 2 < N) { gx += sg[4*t+2];     gy += sg[4*t+3]; }
        // edge e=i: u = p[i+1]-p[i]; grad_p[i] += -s*u (owns str energy)
        if (i <= N) {
            float ux = sp[2*(t+3)+0] - sp[2*(t+2)+0];
            float uy = sp[2*(t+3)+1] - sp[2*(t+2)+1];
            float len  = sqrtf(ux * ux + uy * uy);
            float dlen = len - rest[i];
            float sfac = kstretch * dlen / len;
            gx -= sfac * ux; gy -= sfac * uy;
            str_p = 0.5f * kstretch * dlen * dlen;
        }
        // edge e=i-1: u = p[i]-p[i-1]; grad_p[i] += +s*u
        if (i >= 1) {
            float ux = sp[2*(t+2)+0] - sp[2*(t+1)+0];
            float uy = sp[2*(t+2)+1] - sp[2*(t+1)+1];
            float len  = sqrtf(ux * ux + uy * uy);
            float dlen = len - rest[i - 1];
            float sfac = kstretch * dlen / len;
            gx += sfac * ux; gy += sfac * uy;
        }
        float ox = (i < 2) ? 0.0f : -gx;   // first 4 flat coords fixed
        float oy = (i < 2) ? 0.0f : -gy;
        out[3 + 2*i + 0] = ox;
        out[3 + 2*i + 1] = oy;
    }

    // ---- Phase 3: wave32 shuffle reduction, then per-block (fixed order).
    float rv = rot_p, sv = str_p;
    #pragma unroll
    for (int off = 16; off > 0; off >>= 1) {
        rv += __shfl_down(rv, off, 32);
        sv += __shfl_down(sv, off, 32);
    }
    const int wid  = t >> 5;
    const int lane = t & 31;
    if (lane == 0) { swr[wid] = rv; sws[wid] = sv; }
    __syncthreads();
    if (t == 0) {
        float rb = 0.0f, sb = 0.0f;
        #pragma unroll
        for (int w = 0; w < NWAVES; ++w) { rb += swr[w]; sb += sws[w]; }
        if (blockpart) {
            blockpart[2 * blockIdx.x + 0] = rb;
            blockpart[2 * blockIdx.x + 1] = sb;
        } else {
            atomicAdd(&out[0], rb + sb);
            atomicAdd(&out[1], rb);
            atomicAdd(&out[2], sb);
        }
    }
}

// ---------------------------------------------------------------------------
// Deterministic final reduction of per-block partials (fixed order).
// ---------------------------------------------------------------------------
__global__ __launch_bounds__(BLK) void final_reduce_kernel(
    const float* __restrict__ part, int nblocks, float* __restrict__ out)
{
    __shared__ float swr[NWAVES];
    __shared__ float sws[NWAVES];
    float r = 0.0f, s = 0.0f;
    for (int i = threadIdx.x; i < nblocks; i += BLK) {
        r += part[2 * i + 0];
        s += part[2 * i + 1];
    }
    #pragma unroll
    for (int off = 16; off > 0; off >>= 1) {
        r += __shfl_down(r, off, 32);
        s += __shfl_down(s, off, 32);
    }
    const int wid  = threadIdx.x >> 5;
    const int lane = threadIdx.x & 31;
    if (lane == 0) { swr[wid] = r; sws[wid] = s; }
    __syncthreads();
    if (threadIdx.x == 0) {
        float rb = 0.0f, sb = 0.0f;
        #pragma unroll
        for (int w = 0; w < NWAVES; ++w) { rb += swr[w]; sb += sws[w]; }
        out[1] = rb;
        out[2] = sb;
        out[0] = rb + sb;
    }
}

__global__ void zero3_kernel(float* out) {
    if (threadIdx.x < 3) out[threadIdx.x] = 0.0f;
}

extern "C" void kernel_launch(void* const* d_in, const int* in_sizes, int n_in,
                              void* d_out, int out_size, void* d_ws, size_t ws_size,
                              hipStream_t stream) {
    const float* pos      = (const float*)d_in[0];
    const float* th_ss    = (const float*)d_in[1];
    const float* rest     = (const float*)d_in[2];
    const float* kstiff   = (const float*)d_in[3];
    const float* ksoft    = (const float*)d_in[4];
    const float* kstretch = (const float*)d_in[5];
    const int2*  buckle   = (const int2*)d_in[6];
    float* out = (float*)d_out;

    const int NP = in_sizes[0] / 2;   // points = N+2
    const int N  = NP - 2;            // hinges
    const int nblocks = (NP + BLK - 1) / BLK;

    const bool use_ws = ws_size >= (size_t)nblocks * 2 * sizeof(float);
    float* part = use_ws ? (float*)d_ws : nullptr;

    if (!use_ws) zero3_kernel<<<1, 64, 0, stream>>>(out);   // atomic fallback

    chain_force_kernel<<<nblocks, BLK, 0, stream>>>(
        pos, th_ss, rest, kstiff, ksoft, kstretch, buckle, out, part, N);

    if (use_ws)
        final_reduce_kernel<<<1, BLK, 0, stream>>>(part, nblocks, out);
}